// LogicGateLayer_42288247996562
// MI455X (gfx1250) — compile-verified
//
#include <hip/hip_runtime.h>

typedef __attribute__((ext_vector_type(2))) float v2f;
typedef __attribute__((ext_vector_type(4))) float v4f;
typedef __attribute__((ext_vector_type(8))) float v8f;
typedef __attribute__((ext_vector_type(4))) int   v4i;

#define NUM_NEURONS 32768
#define INPUT_SIZE  8192
#define BATCH       512

// Coefficient matrix rows packed as signed nibbles (value range -2..2):
// op_k(a,b) = C[k][0] + C[k][1]*a + C[k][2]*b + C[k][3]*ab
// nibble j (bits 4j+3:4j) = two's-complement C[k][j].
__host__ __device__ constexpr unsigned CrowPk(int k) {
    constexpr unsigned rows[16] = {
        0x0000u, // 0: zero          ( 0, 0, 0, 0)
        0x1000u, // 1: ab            ( 0, 0, 0, 1)
        0xF010u, // 2: a - ab        ( 0, 1, 0,-1)
        0x0010u, // 3: a             ( 0, 1, 0, 0)
        0xF100u, // 4: b - ab        ( 0, 0, 1,-1)
        0x0100u, // 5: b             ( 0, 0, 1, 0)
        0xE110u, // 6: a+b-2ab       ( 0, 1, 1,-2)
        0xF110u, // 7: a+b-ab        ( 0, 1, 1,-1)
        0x1FF1u, // 8: 1-(a+b-ab)    ( 1,-1,-1, 1)
        0x2FF1u, // 9: 1-(a+b-2ab)   ( 1,-1,-1, 2)
        0x0F01u, // 10: 1-b          ( 1, 0,-1, 0)
        0x1F01u, // 11: 1-b+ab       ( 1, 0,-1, 1)
        0x00F1u, // 12: 1-a          ( 1,-1, 0, 0)
        0x10F1u, // 13: 1-a+ab       ( 1,-1, 0, 1)
        0xF001u, // 14: 1-ab         ( 1, 0, 0,-1)
        0x0001u  // 15: one          ( 1, 0, 0, 0)
    };
    return rows[k];
}

// Extract signed-nibble j from packed row (j is runtime, uniform per lane).
__device__ __forceinline__ float nib(unsigned row, int j) {
    int v = (int)((row >> (4 * j)) & 0xFu);
    v = (v ^ 8) - 8;                 // sign-extend 4-bit -> v_bfe_i32
    return (float)v;                 // v_cvt_f32_i32
}

// Stage 1: coeffs[n][0:4] = softmax(gw[n][0:16]) @ C  via V_WMMA_F32_16X16X4_F32.
// One wave32 per 16-neuron tile; K=16 split into 4 chunks of K=4.
// NOTE: weights kept in 16 named scalars (NOT an array) so `laneLo ? wA : wB`
// stays a single v_cndmask instead of a dynamic-index select chain.
__global__ void lgl_coeff_wmma(const float* __restrict__ gw,
                               float* __restrict__ coeffs) {
    const int  lane     = threadIdx.x & 31;
    const int  wave     = (blockIdx.x * blockDim.x + threadIdx.x) >> 5;
    const int  tileBase = wave * 16;               // first neuron of this tile
    const int  neuron   = tileBase + (lane & 15);  // lanes L and L+16 share neuron
    const bool laneLo   = (lane < 16);
    const int  ncol     = lane & 15;
    const bool colLive  = (ncol < 4);

    // Load this neuron's 16 gate weights (redundant x2 across half-waves; cheap).
    const v4f* g4 = (const v4f*)(gw + (size_t)neuron * 16);
    const v4f t0 = g4[0], t1 = g4[1], t2 = g4[2], t3 = g4[3];
    float w0  = t0.x, w1  = t0.y, w2  = t0.z, w3  = t0.w;
    float w4  = t1.x, w5  = t1.y, w6  = t1.z, w7  = t1.w;
    float w8  = t2.x, w9  = t2.y, w10 = t2.z, w11 = t2.w;
    float w12 = t3.x, w13 = t3.y, w14 = t3.z, w15 = t3.w;

    // Softmax over the 16 ops (TAU = 1): max-reduce tree, exp, sum, normalize.
    float m = fmaxf(fmaxf(fmaxf(fmaxf(w0, w1), fmaxf(w2, w3)),
                          fmaxf(fmaxf(w4, w5), fmaxf(w6, w7))),
                    fmaxf(fmaxf(fmaxf(w8, w9), fmaxf(w10, w11)),
                          fmaxf(fmaxf(w12, w13), fmaxf(w14, w15))));
    w0  = __expf(w0  - m); w1  = __expf(w1  - m);
    w2  = __expf(w2  - m); w3  = __expf(w3  - m);
    w4  = __expf(w4  - m); w5  = __expf(w5  - m);
    w6  = __expf(w6  - m); w7  = __expf(w7  - m);
    w8  = __expf(w8  - m); w9  = __expf(w9  - m);
    w10 = __expf(w10 - m); w11 = __expf(w11 - m);
    w12 = __expf(w12 - m); w13 = __expf(w13 - m);
    w14 = __expf(w14 - m); w15 = __expf(w15 - m);
    const float s = ((w0 + w1) + (w2 + w3)) + ((w4 + w5) + (w6 + w7))
                  + ((w8 + w9) + (w10 + w11)) + ((w12 + w13) + (w14 + w15));
    const float inv = 1.0f / s;
    w0  *= inv; w1  *= inv; w2  *= inv; w3  *= inv;
    w4  *= inv; w5  *= inv; w6  *= inv; w7  *= inv;
    w8  *= inv; w9  *= inv; w10 *= inv; w11 *= inv;
    w12 *= inv; w13 *= inv; w14 *= inv; w15 *= inv;

    // A-frag (16x4 f32, 2 VGPRs): lane<16 -> K = 4q+{0,1}; lane>=16 -> K = 4q+{2,3}.
    // B-frag (4x16 f32, 2 VGPRs): mirrored; columns >= 4 are zero padding.
    v8f acc = {};
    {   // q = 0 (K = 0..3)
        v2f A, B;
        A.x = laneLo ? w0 : w2;
        A.y = laneLo ? w1 : w3;
        const unsigned r0 = laneLo ? CrowPk(0) : CrowPk(2);
        const unsigned r1 = laneLo ? CrowPk(1) : CrowPk(3);
        B.x = colLive ? nib(r0, ncol) : 0.0f;
        B.y = colLive ? nib(r1, ncol) : 0.0f;
        acc = __builtin_amdgcn_wmma_f32_16x16x4_f32(false, A, false, B,
                                                    (short)0, acc, false, false);
    }
    {   // q = 1 (K = 4..7)
        v2f A, B;
        A.x = laneLo ? w4 : w6;
        A.y = laneLo ? w5 : w7;
        const unsigned r0 = laneLo ? CrowPk(4) : CrowPk(6);
        const unsigned r1 = laneLo ? CrowPk(5) : CrowPk(7);
        B.x = colLive ? nib(r0, ncol) : 0.0f;
        B.y = colLive ? nib(r1, ncol) : 0.0f;
        acc = __builtin_amdgcn_wmma_f32_16x16x4_f32(false, A, false, B,
                                                    (short)0, acc, false, false);
    }
    {   // q = 2 (K = 8..11)
        v2f A, B;
        A.x = laneLo ? w8 : w10;
        A.y = laneLo ? w9 : w11;
        const unsigned r0 = laneLo ? CrowPk(8) : CrowPk(10);
        const unsigned r1 = laneLo ? CrowPk(9) : CrowPk(11);
        B.x = colLive ? nib(r0, ncol) : 0.0f;
        B.y = colLive ? nib(r1, ncol) : 0.0f;
        acc = __builtin_amdgcn_wmma_f32_16x16x4_f32(false, A, false, B,
                                                    (short)0, acc, false, false);
    }
    {   // q = 3 (K = 12..15)
        v2f A, B;
        A.x = laneLo ? w12 : w14;
        A.y = laneLo ? w13 : w15;
        const unsigned r0 = laneLo ? CrowPk(12) : CrowPk(14);
        const unsigned r1 = laneLo ? CrowPk(13) : CrowPk(15);
        B.x = colLive ? nib(r0, ncol) : 0.0f;
        B.y = colLive ? nib(r1, ncol) : 0.0f;
        acc = __builtin_amdgcn_wmma_f32_16x16x4_f32(false, A, false, B,
                                                    (short)0, acc, false, false);
    }

    // D layout (16x16 f32): VGPR r -> (M=r, N=lane) lanes 0-15, (M=r+8, N=lane-16)
    // lanes 16-31. Only N=0..3 carries real data.
    if (colLive) {
        const int mOff = laneLo ? 0 : 8;
        #pragma unroll
        for (int r = 0; r < 8; ++r)
            coeffs[(size_t)(tileBase + mOff + r) * 4 + ncol] = acc[r];
    }
}

// Stage 2: out[b,n] = c0 + c1*a + c2*b + c3*a*b with a,b gathered from x's row.
// Thread = 4 consecutive neurons; loops over 8 batch rows reusing idx/coeffs in
// registers. x (16 MB) is L2-resident; output streamed with nontemporal stores.
__global__ void __launch_bounds__(256)
lgl_main(const float* __restrict__ x,
         const int*   __restrict__ idx,
         const float* __restrict__ coeffs,
         float*       __restrict__ out) {
    const int nBase = (blockIdx.x * blockDim.x + threadIdx.x) * 4;
    const int bBase = blockIdx.y * 8;

    // idx pairs for 4 neurons: two B128 loads, coalesced.
    const v4i p01 = ((const v4i*)idx)[(nBase >> 1) + 0]; // neurons nBase, nBase+1
    const v4i p23 = ((const v4i*)idx)[(nBase >> 1) + 1]; // neurons nBase+2, nBase+3
    // coefficients for 4 neurons: four B128 loads, coalesced.
    const v4f c0 = ((const v4f*)coeffs)[nBase + 0];
    const v4f c1 = ((const v4f*)coeffs)[nBase + 1];
    const v4f c2 = ((const v4f*)coeffs)[nBase + 2];
    const v4f c3 = ((const v4f*)coeffs)[nBase + 3];

    #pragma unroll
    for (int bb = 0; bb < 8; ++bb) {
        const int b = bBase + bb;
        const float* __restrict__ xr = x + (size_t)b * INPUT_SIZE;

        const float a0 = xr[p01.x], b0 = xr[p01.y];
        const float a1 = xr[p01.z], b1 = xr[p01.w];
        const float a2 = xr[p23.x], b2 = xr[p23.y];
        const float a3 = xr[p23.z], b3 = xr[p23.w];

        v4f o;
        o.x = fmaf(c0.w, a0 * b0, fmaf(c0.z, b0, fmaf(c0.y, a0, c0.x)));
        o.y = fmaf(c1.w, a1 * b1, fmaf(c1.z, b1, fmaf(c1.y, a1, c1.x)));
        o.z = fmaf(c2.w, a2 * b2, fmaf(c2.z, b2, fmaf(c2.y, a2, c2.x)));
        o.w = fmaf(c3.w, a3 * b3, fmaf(c3.z, b3, fmaf(c3.y, a3, c3.x)));

        // 64 MB write-once stream: NT hint keeps x/idx/coeffs resident in L2.
        v4f* dst = (v4f*)(out + (size_t)b * NUM_NEURONS + nBase);
        __builtin_nontemporal_store(o, dst);
    }
}

extern "C" void kernel_launch(void* const* d_in, const int* in_sizes, int n_in,
                              void* d_out, int out_size, void* d_ws, size_t ws_size,
                              hipStream_t stream) {
    const float* x      = (const float*)d_in[0]; // (512, 8192) f32
    const float* gw     = (const float*)d_in[1]; // (32768, 16) f32
    const int*   idx    = (const int*)  d_in[2]; // (32768, 2) i32
    float*       out    = (float*)d_out;         // (512, 32768) f32
    float*       coeffs = (float*)d_ws;          // 32768 * 4 floats = 512 KB scratch

    // Stage 1: 32768/16 = 2048 wave-tiles; 8 waves (256 threads) per block.
    const int waves   = NUM_NEURONS / 16;
    const int blocks1 = waves / 8;               // 256 blocks
    lgl_coeff_wmma<<<blocks1, 256, 0, stream>>>(gw, coeffs);

    // Stage 2: x-dim covers neurons (32768 / (256*4) = 32), y-dim batch groups (512/8 = 64).
    dim3 grid(NUM_NEURONS / (256 * 4), BATCH / 8);
    lgl_main<<<grid, 256, 0, stream>>>(x, idx, coeffs, out);
}